// HLF_14474039787790
// MI455X (gfx1250) — compile-verified
//
#include <hip/hip_runtime.h>

typedef __attribute__((ext_vector_type(2))) float v2f;
typedef __attribute__((ext_vector_type(4))) float v4f;
typedef __attribute__((ext_vector_type(8))) float v8f;

#define BATCH   128
#define K_DIM   2048
#define N_OUT   14
#define ROW_LEN 49          // 7*7 spatial
#define ROWS_PER_BLOCK 128  // rows of (B*2048) handled per block

// ---------------------------------------------------------------------------
// Kernel 1 (bandwidth-bound part): feats[r] = (2*sum49(x1[r]) + sum49(x2[r]))/49
// Softmax over a singleton axis == 1, so h = 2*x1 + x2 and x3 is never read.
// Strategy: each block streams 128 rows (6272 floats = 25088 B, 16B aligned)
// from x1 and x2 with coalesced b128 loads, fuses 2*a+b into LDS, then 128
// threads each sum 49 consecutive LDS floats (stride 49 is odd -> gcd(49,64)=1
// -> conflict-free across the 64 LDS banks).
// ---------------------------------------------------------------------------
__global__ __launch_bounds__(256)
void HLF_feats_kernel(const float* __restrict__ x1,
                      const float* __restrict__ x2,
                      float* __restrict__ feats) {
    __shared__ float lds[ROWS_PER_BLOCK * ROW_LEN];  // 25088 bytes

    const int tid = threadIdx.x;
    const long long blockBase = (long long)blockIdx.x * (ROWS_PER_BLOCK * ROW_LEN);
    const v4f* __restrict__ x1v = (const v4f*)(x1 + blockBase);
    const v4f* __restrict__ x2v = (const v4f*)(x2 + blockBase);

    const int nvec = ROWS_PER_BLOCK * ROW_LEN / 4;   // 1568 float4 per block
    for (int i = tid; i < nvec; i += 256) {
        v4f a = x1v[i];   // global_load_b128, fully coalesced
        v4f b = x2v[i];
        v4f h;
        h.x = 2.0f * a.x + b.x;
        h.y = 2.0f * a.y + b.y;
        h.z = 2.0f * a.z + b.z;
        h.w = 2.0f * a.w + b.w;
        *(v4f*)(&lds[i * 4]) = h;                    // ds_store_b128
    }
    __syncthreads();

    if (tid < ROWS_PER_BLOCK) {
        float s = 0.0f;
#pragma unroll
        for (int k = 0; k < ROW_LEN; ++k)
            s += lds[tid * ROW_LEN + k];
        feats[(long long)blockIdx.x * ROWS_PER_BLOCK + tid] = s * (1.0f / 49.0f);
    }
}

// ---------------------------------------------------------------------------
// Kernel 2: logits(128x14) = feats(128x2048) @ W^T(2048x14) + b
// via V_WMMA_F32_16X16X4_F32. One workgroup, 8 waves, each wave owns one
// 16-row M-tile; N padded 14 -> 16 with zero columns; K in steps of 4
// -> 512 wmma ops per wave, f32 end to end (matches reference precision).
//
// Fragment layouts (CDNA5 ISA 7.12.2, 32-bit operands, wave32):
//   A 16x4 : lane l, vgpr v  ->  A[l%16][2*(l/16) + v]
//   B 4x16 : lane l, vgpr v  ->  B[2*(l/16) + v][l%16]
//   C 16x16: lane l, vgpr v  ->  C[v + 8*(l/16)][l%16]
// ---------------------------------------------------------------------------
__global__ __launch_bounds__(256)
void HLF_logits_kernel(const float* __restrict__ feats,
                       const float* __restrict__ W,
                       const float* __restrict__ bias,
                       float* __restrict__ logits) {
    const int tid  = threadIdx.x;
    const int lane = tid & 31;
    const int wave = tid >> 5;        // 0..7 -> M-tile
    const int n    = lane & 15;       // output column (padded to 16)
    const int half = lane >> 4;       // 0 or 1
    const bool nvalid = (n < N_OUT);

    const int m = wave * 16 + n;      // A-fragment row for this lane
    const float* __restrict__ arow = feats + (long long)m * K_DIM + 2 * half;
    const float* __restrict__ brow = W + (long long)(nvalid ? n : 0) * K_DIM + 2 * half;

    v8f c = {};
    for (int k0 = 0; k0 < K_DIM; k0 += 4) {
        v2f a = *(const v2f*)(arow + k0);          // A[m][k0+2h], A[m][k0+2h+1]
        v2f b = *(const v2f*)(brow + k0);          // W[n][k0+2h], W[n][k0+2h+1]
        if (!nvalid) { b.x = 0.0f; b.y = 0.0f; }   // zero-pad columns 14,15
        // (neg_a, A, neg_b, B, c_mod, C, reuse_a, reuse_b)
        c = __builtin_amdgcn_wmma_f32_16x16x4_f32(
                false, a, false, b, (short)0, c, false, false);
    }

    if (nvalid) {
        const float bv = bias[n];
#pragma unroll
        for (int v = 0; v < 8; ++v) {
            const int M = wave * 16 + v + half * 8;
            logits[M * N_OUT + n] = c[v] + bv;
        }
    }
}

extern "C" void kernel_launch(void* const* d_in, const int* in_sizes, int n_in,
                              void* d_out, int out_size, void* d_ws, size_t ws_size,
                              hipStream_t stream) {
    const float* x1 = (const float*)d_in[0];
    const float* x2 = (const float*)d_in[1];
    // d_in[2] = x3: mathematically unused (softmax over singleton axis == 1)
    const float* W  = (const float*)d_in[3];
    const float* b  = (const float*)d_in[4];

    float* logits = (float*)d_out;                 // first 128*14 floats
    float* feats  = (float*)d_out + BATCH * N_OUT; // next 128*2048 floats

    const int totalRows = BATCH * K_DIM;           // 262144
    HLF_feats_kernel<<<dim3(totalRows / ROWS_PER_BLOCK), 256, 0, stream>>>(x1, x2, feats);
    HLF_logits_kernel<<<dim3(1), 256, 0, stream>>>(feats, W, b, logits);
}